// _QuantumGateModule_65481071404062
// MI455X (gfx1250) — compile-verified
//
#include <hip/hip_runtime.h>

// out[b,q] = cos(x[b,q] + theta[q])   (analytic collapse of the RX circuit)
// computed as the exact f32 GEMM  out = F @ G  with
//   F[b, 2w+t] = (t==0) ? cos(x[b,w]) : sin(x[b,w])        (2048 x 32, K padded)
//   G[2w+t, q] = (q==w) ? ((t==0) ? cos(th_w) : -sin(th_w)) : 0   (32 x 16, N padded)
// using 8 chained V_WMMA_F32_16X16X4_F32 per 16-batch tile.

typedef float v2f __attribute__((ext_vector_type(2)));
typedef float v8f __attribute__((ext_vector_type(8)));

#define NQ    14
#define BATCHN 2048

__global__ __launch_bounds__(256)
void qrx_expz_wmma(const float* __restrict__ x,
                   const float* __restrict__ theta,
                   float* __restrict__ out)
{
    const int lane = threadIdx.x & 31;
    const int wave = threadIdx.x >> 5;
    const int tile = blockIdx.x * 8 + wave;   // one 16-batch tile per wave
    const int b0   = tile * 16;
    const int half = lane >> 4;               // 0: lanes 0-15, 1: lanes 16-31
    const int m    = lane & 15;               // A-row (batch-in-tile) / B-col (qubit)
    const int b    = b0 + m;

    // B diagonal entries live only on lane column q == m: need cos/sin(theta[m]).
    float ct = 0.0f, st = 0.0f;
    if (m < NQ) {
        const float tv = theta[m];
        ct = __builtin_cosf(tv);
        st = __builtin_sinf(tv);
    }

    v8f acc = {};   // C/D 16x16 f32 accumulator (8 VGPRs)

#pragma unroll
    for (int s = 0; s < 8; ++s) {
        // This WMMA step covers K = 4s..4s+3.
        // A layout (16x4 f32): v0 = K = 4s+2*half, v1 = K = 4s+2*half+1, row M = m.
        // K = 2w + t  =>  this lane-half supplies qubit w = 2s + half, t = {0,1}.
        const int w = 2 * s + half;

        v2f a;
        if (w < NQ) {
            const float xv = x[b * NQ + w];
            a.x = __builtin_cosf(xv);     // t=0
            a.y = __builtin_sinf(xv);     // t=1
        } else {
            a.x = 0.0f; a.y = 0.0f;       // K padding (28..31)
        }

        // B layout (4x16 f32): v0 = row K = 4s+2*half, v1 = row K+1; col N = m.
        // G is diagonal in (w,q): nonzero only where m == w.
        const bool diag = (m == w) && (w < NQ);
        v2f bm;
        bm.x = diag ?  ct : 0.0f;         // t=0 row: cos(theta_w)
        bm.y = diag ? -st : 0.0f;         // t=1 row: -sin(theta_w)

        // D = A x B + C, full f32 path (no precision loss vs reference).
        acc = __builtin_amdgcn_wmma_f32_16x16x4_f32(
                  /*neg_a=*/false, a,
                  /*neg_b=*/false, bm,
                  /*c_mod=*/(short)0, acc,
                  /*reuse_a=*/false, /*reuse_b=*/false);
    }

    // D layout: VGPR r holds out[b0 + r + 8*half, q = m].
    if (m < NQ) {
#pragma unroll
        for (int r = 0; r < 8; ++r) {
            out[(b0 + r + 8 * half) * NQ + m] = acc[r];
        }
    }
}

extern "C" void kernel_launch(void* const* d_in, const int* in_sizes, int n_in,
                              void* d_out, int out_size, void* d_ws, size_t ws_size,
                              hipStream_t stream)
{
    const float* x     = (const float*)d_in[0];   // (2048, 14) f32
    const float* theta = (const float*)d_in[1];   // (14,)      f32
    float*       out   = (float*)d_out;           // (2048, 14) f32

    // 8 waves/block, 1 tile(16 batches)/wave -> 128 tiles -> 16 blocks.
    dim3 grid(BATCHN / (16 * 8));
    dim3 block(256);
    hipLaunchKernelGGL(qrx_expz_wmma, grid, block, 0, stream, x, theta, out);
}